// IMLayer_22368189677863
// MI455X (gfx1250) — compile-verified
//
#include <hip/hip_runtime.h>
#include <math.h>

// ---------------------------------------------------------------------------
// SSM layer for MI455X (gfx1250, wave32, WMMA).
//   P=H=256, L=4096, BATCH=4  ->  ROWS = 16384 flattened (b,l) rows.
// Pipeline:
//   0) pack:       B_c / C_c -> WMMA-fragment-native, re/im deinterleaved
//   1) params:     per-p M(2x2), F-scales, and M^64 via repeated squaring
//   2) gemm_bu:    Bu(re,im) = u @ B_c^T      (V_WMMA_F32_16X16X4_F32, LDS B)
//   3) scan_chunk: per-(b,p,chunk) local recurrence partials (zero init)
//   4) scan_carry: cross-chunk scan with constant M^64, emits carry-in states
//   5) scan_emit:  re-run chunks with carry-in, overwrite Bu buffers with ys
//   6) gemm_out:   out = ys_re@Cre^T - ys_im@Cim^T + D*u   (WMMA f32, LDS C)
// Workspace: ~2MB packed weights/params + 32MB Bu/ys + 2MB scan state ~= 36MB.
// ---------------------------------------------------------------------------

#define PDIM   256
#define HDIM   256
#define LDIM   4096
#define NB     4
#define ROWS   (NB * LDIM)        // 16384
#define CHUNK  64
#define NCHUNK (LDIM / CHUNK)     // 64
#define KSTEPS 64                 // 256 / 4
#define TILEF  8192               // floats per packed 16-col tile (64*2*64)

typedef __attribute__((ext_vector_type(2))) float v2f;
typedef __attribute__((ext_vector_type(8))) float v8f;
typedef __attribute__((ext_vector_type(4))) int   v4i;

// f32 WMMA: D(16x16) = A(16x4) * B(4x16) + C
// 8-arg form: (neg_a, A, neg_b, B, c_mod, C, reuse_a, reuse_b)
__device__ __forceinline__ v8f wmma_f32(v2f a, v2f b, v8f c) {
    return __builtin_amdgcn_wmma_f32_16x16x4_f32(false, a, false, b,
                                                 (short)0, c, false, false);
}

// gfx1250 async global->LDS copy path (ASYNCcnt-tracked), with fallback.
// Probe-confirmed signature: (v4i AS(1)* src, v4i AS(3)* dst, Ii off, Ii cpol)
#if __has_builtin(__builtin_amdgcn_global_load_async_to_lds_b128) && \
    __has_builtin(__builtin_amdgcn_s_wait_asynccnt)
#define USE_ASYNC_LDS 1
#else
#define USE_ASYNC_LDS 0
#endif

// Stage one packed 32KB fragment tile (TILEF floats) into LDS; 256 threads.
__device__ __forceinline__ void stage_tile(float* lds, const float* g) {
#if USE_ASYNC_LDS
#pragma unroll
    for (int i = 0; i < TILEF / (256 * 4); ++i) {
        int off = (i * 256 + (int)threadIdx.x) * 4;   // 16B per lane per op
        __builtin_amdgcn_global_load_async_to_lds_b128(
            (__attribute__((address_space(1))) v4i*)(g + off),
            (__attribute__((address_space(3))) v4i*)(lds + off),
            0, 0);
    }
    __builtin_amdgcn_s_wait_asynccnt(0);
    __syncthreads();
#else
#pragma unroll
    for (int i = 0; i < TILEF / (256 * 4); ++i) {
        int off = (i * 256 + (int)threadIdx.x) * 4;
        *(float4*)(lds + off) = *(const float4*)(g + off);
    }
    __syncthreads();
#endif
}

// ---------------------------------------------------------------------------
// Kernel 0: pack a complex weight tensor (N=256, K=256, 2) into fragment-
// native layout: dst[ct][ks][comp][lane*2+v], ct = 16-col tile, ks = K/4 step.
// Lane mapping matches the f32 WMMA B fragment: n = lane%16, K = ks*4 +
// 2*(lane/16) + v.  Total 131072 floats; pure gather, launched once per tensor.
// ---------------------------------------------------------------------------
__global__ __launch_bounds__(256)
void ssm_pack_kernel(const float* __restrict__ src, float* __restrict__ dst) {
    int idx  = blockIdx.x * 256 + threadIdx.x;  // 0 .. 131071
    int v    = idx & 1;
    int l    = (idx >> 1) & 31;
    int comp = (idx >> 6) & 1;
    int ks   = (idx >> 7) & 63;
    int ct   = idx >> 13;
    int n    = l & 15;
    int kh   = l >> 4;
    int k    = ks * 4 + kh * 2 + v;
    dst[idx] = src[((size_t)(ct * 16 + n) * 256 + k) * 2 + comp];
}

// ---------------------------------------------------------------------------
// Kernel 1: per-state parameters.  prm layout: prm[i*PDIM + p], i in 0..9:
//   0..3: M11,M12,M21,M22   4: f1=M11*dts   5: f2=M21*dts   6..9: M^64
// ---------------------------------------------------------------------------
__global__ __launch_bounds__(PDIM)
void ssm_params_kernel(const float* __restrict__ A_diag,
                       const float* __restrict__ dt,
                       float* __restrict__ prm) {
    int p = threadIdx.x;
    float dts  = 1.0f / (1.0f + __expf(-dt[p]));
    float Ad   = fmaxf(A_diag[p], 0.0f);
    float S    = 1.0f + dts * dts * Ad;
    float invS = 1.0f / S;
    float M11  = 1.0f - dts * dts * Ad * invS;
    float M12  = -dts * Ad * invS;
    float M21  = dts * invS;
    float M22  = invS;

    // M^64 by 6 squarings (M is time-invariant => chunk matrix is constant).
    float a = M11, b = M12, c = M21, e = M22;
#pragma unroll
    for (int i = 0; i < 6; ++i) {
        float na = a * a + b * c;
        float nb = a * b + b * e;
        float nc = c * a + e * c;
        float ne = c * b + e * e;
        a = na; b = nb; c = nc; e = ne;
    }
    prm[0 * PDIM + p] = M11;
    prm[1 * PDIM + p] = M12;
    prm[2 * PDIM + p] = M21;
    prm[3 * PDIM + p] = M22;
    prm[4 * PDIM + p] = M11 * dts;
    prm[5 * PDIM + p] = M21 * dts;
    prm[6 * PDIM + p] = a;
    prm[7 * PDIM + p] = b;
    prm[8 * PDIM + p] = c;
    prm[9 * PDIM + p] = e;
}

// ---------------------------------------------------------------------------
// Kernel 2: Bu = u @ B_c^T.  8 waves/block, each wave owns a 16x16 tile;
// all 8 waves share one 16-col B tile staged in LDS (32KB, async path).
// Inner loop per K-step: 1 b64 A load + 2 ds b64 fragment loads + 2 WMMAs.
// ---------------------------------------------------------------------------
__global__ __launch_bounds__(256)
void ssm_gemm_bu_kernel(const float* __restrict__ u,     // [ROWS][HDIM]
                        const float* __restrict__ Bpk,   // packed fragments
                        float* __restrict__ bu_re,       // [ROWS][PDIM]
                        float* __restrict__ bu_im) {
    __shared__ float lbs[TILEF];
    const int wave = threadIdx.x >> 5;
    const int lane = threadIdx.x & 31;
    const int m    = lane & 15;
    const int kh   = lane >> 4;
    const int row0 = blockIdx.x * 128 + wave * 16;
    const int col0 = blockIdx.y * 16;

    stage_tile(lbs, Bpk + (size_t)blockIdx.y * TILEF);

    v8f accr = {}, acci = {};
    const float* arow = u + (size_t)(row0 + m) * HDIM + kh * 2;
    const float* lb   = lbs + lane * 2;
#pragma unroll 4
    for (int ks = 0; ks < KSTEPS; ++ks) {
        v2f a   = *(const v2f*)(arow + ks * 4);
        v2f bre = *(const v2f*)(lb + ks * 128);
        v2f bim = *(const v2f*)(lb + ks * 128 + 64);
        accr = wmma_f32(a, bre, accr);               // shares A fragment
        acci = wmma_f32(a, bim, acci);
    }

    // C/D layout: VGPR r -> row r (lanes 0-15) / row r+8 (lanes 16-31), col m
    const int rbase = row0 + kh * 8;
#pragma unroll
    for (int r = 0; r < 8; ++r) {
        size_t idx = (size_t)(rbase + r) * PDIM + col0 + m;
        bu_re[idx] = accr[r];
        bu_im[idx] = acci[r];
    }
}

// ---------------------------------------------------------------------------
// Kernel 3: per-chunk local recurrence with zero initial state.
// One thread per (b, chunk, p); block = all p (coalesced over p).
// ---------------------------------------------------------------------------
__global__ __launch_bounds__(PDIM)
void ssm_scan_chunk_kernel(const float* __restrict__ bu_re,
                           const float* __restrict__ bu_im,
                           const float* __restrict__ prm,
                           float4* __restrict__ chunkpart) {
    const int b = blockIdx.x / NCHUNK;
    const int c = blockIdx.x % NCHUNK;
    const int p = threadIdx.x;
    const float M11 = prm[0 * PDIM + p], M12 = prm[1 * PDIM + p];
    const float M21 = prm[2 * PDIM + p], M22 = prm[3 * PDIM + p];
    const float f1  = prm[4 * PDIM + p], f2  = prm[5 * PDIM + p];

    float x1r = 0.f, x1i = 0.f, x2r = 0.f, x2i = 0.f;
    size_t base = ((size_t)b * LDIM + (size_t)c * CHUNK) * PDIM + p;
    for (int t = 0; t < CHUNK; ++t) {
        float br = bu_re[base + (size_t)t * PDIM];
        float bi = bu_im[base + (size_t)t * PDIM];
        float n1r = fmaf(M11, x1r, fmaf(M12, x2r, f1 * br));
        float n1i = fmaf(M11, x1i, fmaf(M12, x2i, f1 * bi));
        float n2r = fmaf(M21, x1r, fmaf(M22, x2r, f2 * br));
        float n2i = fmaf(M21, x1i, fmaf(M22, x2i, f2 * bi));
        x1r = n1r; x1i = n1i; x2r = n2r; x2i = n2i;
    }
    chunkpart[(size_t)(b * NCHUNK + c) * PDIM + p] =
        make_float4(x1r, x1i, x2r, x2i);
}

// ---------------------------------------------------------------------------
// Kernel 4: cross-chunk scan with the constant chunk matrix M^64.
// Writes the carry-in (state BEFORE each chunk).
// ---------------------------------------------------------------------------
__global__ __launch_bounds__(PDIM)
void ssm_scan_carry_kernel(const float4* __restrict__ chunkpart,
                           const float* __restrict__ prm,
                           float4* __restrict__ carry) {
    const int b = blockIdx.x;
    const int p = threadIdx.x;
    const float c11 = prm[6 * PDIM + p], c12 = prm[7 * PDIM + p];
    const float c21 = prm[8 * PDIM + p], c22 = prm[9 * PDIM + p];

    float x1r = 0.f, x1i = 0.f, x2r = 0.f, x2i = 0.f;
    for (int c = 0; c < NCHUNK; ++c) {
        size_t idx = (size_t)(b * NCHUNK + c) * PDIM + p;
        carry[idx] = make_float4(x1r, x1i, x2r, x2i);
        float4 bp  = chunkpart[idx];
        float n1r = fmaf(c11, x1r, fmaf(c12, x2r, bp.x));
        float n1i = fmaf(c11, x1i, fmaf(c12, x2i, bp.y));
        float n2r = fmaf(c21, x1r, fmaf(c22, x2r, bp.z));
        float n2i = fmaf(c21, x1i, fmaf(c22, x2i, bp.w));
        x1r = n1r; x1i = n1i; x2r = n2r; x2i = n2i;
    }
}

// ---------------------------------------------------------------------------
// Kernel 5: re-run chunks with carry-in; overwrite Bu buffers with ys = x2.
// (Each thread reads-then-writes the same address per step -> safe in place.)
// ---------------------------------------------------------------------------
__global__ __launch_bounds__(PDIM)
void ssm_scan_emit_kernel(float* __restrict__ bu_re,   // in: Bu_re, out: ys_re
                          float* __restrict__ bu_im,   // in: Bu_im, out: ys_im
                          const float* __restrict__ prm,
                          const float4* __restrict__ carry) {
    const int b = blockIdx.x / NCHUNK;
    const int c = blockIdx.x % NCHUNK;
    const int p = threadIdx.x;
    const float M11 = prm[0 * PDIM + p], M12 = prm[1 * PDIM + p];
    const float M21 = prm[2 * PDIM + p], M22 = prm[3 * PDIM + p];
    const float f1  = prm[4 * PDIM + p], f2  = prm[5 * PDIM + p];

    float4 x0 = carry[(size_t)(b * NCHUNK + c) * PDIM + p];
    float x1r = x0.x, x1i = x0.y, x2r = x0.z, x2i = x0.w;

    size_t base = ((size_t)b * LDIM + (size_t)c * CHUNK) * PDIM + p;
    for (int t = 0; t < CHUNK; ++t) {
        size_t idx = base + (size_t)t * PDIM;
        float br = bu_re[idx];
        float bi = bu_im[idx];
        float n1r = fmaf(M11, x1r, fmaf(M12, x2r, f1 * br));
        float n1i = fmaf(M11, x1i, fmaf(M12, x2i, f1 * bi));
        float n2r = fmaf(M21, x1r, fmaf(M22, x2r, f2 * br));
        float n2i = fmaf(M21, x1i, fmaf(M22, x2i, f2 * bi));
        x1r = n1r; x1i = n1i; x2r = n2r; x2i = n2i;
        bu_re[idx] = x2r;          // ys (second state component), real
        bu_im[idx] = x2i;          // ys, imag
    }
}

// ---------------------------------------------------------------------------
// Kernel 6: out = ys_re @ Cre^T - ys_im @ Cim^T + D*u.  Same WMMA tiling,
// C fragment tile staged in LDS.
// ---------------------------------------------------------------------------
__global__ __launch_bounds__(256)
void ssm_gemm_out_kernel(const float* __restrict__ ys_re, // [ROWS][PDIM]
                         const float* __restrict__ ys_im,
                         const float* __restrict__ Cpk,   // packed fragments
                         const float* __restrict__ Dvec,  // [HDIM]
                         const float* __restrict__ u,     // [ROWS][HDIM]
                         float* __restrict__ out) {       // [ROWS][HDIM]
    __shared__ float lcs[TILEF];
    const int wave = threadIdx.x >> 5;
    const int lane = threadIdx.x & 31;
    const int m    = lane & 15;
    const int kh   = lane >> 4;
    const int row0 = blockIdx.x * 128 + wave * 16;
    const int col0 = blockIdx.y * 16;

    stage_tile(lcs, Cpk + (size_t)blockIdx.y * TILEF);

    v8f accR = {}, accI = {};
    const float* ar = ys_re + (size_t)(row0 + m) * PDIM + kh * 2;
    const float* ai = ys_im + (size_t)(row0 + m) * PDIM + kh * 2;
    const float* lc = lcs + lane * 2;
#pragma unroll 4
    for (int ks = 0; ks < KSTEPS; ++ks) {
        v2f a_re = *(const v2f*)(ar + ks * 4);
        v2f a_im = *(const v2f*)(ai + ks * 4);
        v2f bre  = *(const v2f*)(lc + ks * 128);
        v2f bim  = *(const v2f*)(lc + ks * 128 + 64);
        accR = wmma_f32(a_re, bre, accR);
        accI = wmma_f32(a_im, bim, accI);
    }

    const float dcol = Dvec[col0 + m];
    const int rbase = row0 + kh * 8;
#pragma unroll
    for (int r = 0; r < 8; ++r) {
        size_t ridx = (size_t)(rbase + r) * HDIM + col0 + m;
        out[ridx] = accR[r] - accI[r] + dcol * u[ridx];
    }
}

// ---------------------------------------------------------------------------
extern "C" void kernel_launch(void* const* d_in, const int* in_sizes, int n_in,
                              void* d_out, int out_size, void* d_ws,
                              size_t ws_size, hipStream_t stream) {
    const float* u  = (const float*)d_in[0];  // (4,4096,256)
    const float* A  = (const float*)d_in[1];  // (256,)
    const float* Bm = (const float*)d_in[2];  // (256,256,2)
    const float* Cm = (const float*)d_in[3];  // (256,256,2)
    const float* Dv = (const float*)d_in[4];  // (256,)
    const float* dt = (const float*)d_in[5];  // (256,)
    float* out = (float*)d_out;

    // Workspace carving (~36 MB total).
    char*  ws  = (char*)d_ws;
    float* prm = (float*)ws;                                  // 10 KB
    float* Bpk = (float*)(ws + (1 << 16));                    // 512 KB
    float* Cpk = (float*)(ws + (1 << 16) + (1 << 19));        // 512 KB
    float* bu_re = (float*)(ws + (1 << 21));                  // 16 MB
    float* bu_im = bu_re + (size_t)ROWS * PDIM;               // 16 MB
    float4* chunkpart = (float4*)(bu_im + (size_t)ROWS * PDIM); // 1 MB
    float4* carry     = chunkpart + (size_t)NB * NCHUNK * PDIM; // 1 MB

    ssm_params_kernel<<<1, PDIM, 0, stream>>>(A, dt, prm);
    ssm_pack_kernel<<<512, 256, 0, stream>>>(Bm, Bpk);
    ssm_pack_kernel<<<512, 256, 0, stream>>>(Cm, Cpk);
    ssm_gemm_bu_kernel<<<dim3(ROWS / 128, PDIM / 16), 256, 0, stream>>>(
        u, Bpk, bu_re, bu_im);
    ssm_scan_chunk_kernel<<<NB * NCHUNK, PDIM, 0, stream>>>(
        bu_re, bu_im, prm, chunkpart);
    ssm_scan_carry_kernel<<<NB, PDIM, 0, stream>>>(chunkpart, prm, carry);
    ssm_scan_emit_kernel<<<NB * NCHUNK, PDIM, 0, stream>>>(
        bu_re, bu_im, prm, carry);
    ssm_gemm_out_kernel<<<dim3(ROWS / 128, HDIM / 16), 256, 0, stream>>>(
        bu_re, bu_im, Cpk, Dv, u, out);
}